// PositionalEncoderGrid_65635690217857
// MI455X (gfx1250) — compile-verified
//
#include <hip/hip_runtime.h>
#include <cstdint>
#include <cstddef>

typedef float v2f __attribute__((ext_vector_type(2)));

namespace {
constexpr int      kL    = 16;
constexpr unsigned kT    = 1u << 19;      // hash table rows per level
constexpr unsigned kMask = kT - 1u;       // T is a power of two
constexpr int      kB    = 524288;        // BATCH
constexpr unsigned kP1   = 2654435761u;
constexpr unsigned kP2   = 805459861u;
}

// Packed per-level (n, 1/n): NS[i] = int(16 * 2^(i/3)); widths = 1/NS in f32.
// Single b64 load per thread instead of two b32 loads.
__constant__ v2f c_nw[16] = {
    {16.f,  1.f/16.f},  {20.f,  1.f/20.f},  {25.f,  1.f/25.f},  {32.f,  1.f/32.f},
    {40.f,  1.f/40.f},  {50.f,  1.f/50.f},  {64.f,  1.f/64.f},  {80.f,  1.f/80.f},
    {101.f, 1.f/101.f}, {128.f, 1.f/128.f}, {161.f, 1.f/161.f}, {203.f, 1.f/203.f},
    {256.f, 1.f/256.f}, {322.f, 1.f/322.f}, {406.f, 1.f/406.f}, {512.f, 1.f/512.f}};

__global__ __launch_bounds__(256) void hashgrid_encode_kernel(
    const float* __restrict__ pos,   // [B,3]
    const float* __restrict__ tab,   // [L,T,2]
    float*       __restrict__ out)   // [B,L*2]
{
    __shared__ float s_pos[48];      // 16 points * 3 coords staged per block
    const int tid = threadIdx.x;

    // ---- Async-stage this block's 16 positions (48 floats) into LDS ----
    // LDS byte address comes from the pointer itself (low 32 bits of a flat
    // LDS pointer == 0-based LDS address); the array pointer is also passed
    // as an asm operand so s_pos escapes and the memory clobber forces the
    // compiler to assume the asm writes it (prevents undef-folding/DCE).
    const float* gsrc = pos + (size_t)blockIdx.x * 48u;
    if (tid < 48) {
        unsigned lds_addr = (unsigned)(uintptr_t)&s_pos[tid];
        unsigned goff     = (unsigned)(tid * 4);
        asm volatile("global_load_async_to_lds_b32 %0, %1, %2 ;; writes LDS at %3"
                     :: "v"(lds_addr), "v"(goff), "s"(gsrc), "v"((void*)s_pos)
                     : "memory");
    }
    asm volatile("s_wait_asynccnt 0" ::: "memory");
    __syncthreads();

    const unsigned gid = blockIdx.x * 256u + (unsigned)tid;
    const unsigned b   = gid >> 4;        // point index
    const unsigned l   = gid & 15u;       // level index
    const int      bl  = tid >> 4;        // point index within block

    const v2f   nw = c_nw[l];
    const float nf = nw.x;                // resolution n
    const float wd = nw.y;                // width = 1/n (f32, matches reference)

    const unsigned primes[3] = {1u, kP1, kP2};
    float    pr[3], om[3];
    unsigned h0[3], h1[3];
    #pragma unroll
    for (int d = 0; d < 3; ++d) {
        float x  = (s_pos[bl * 3 + d] + 3.0f) * (1.0f / 6.0f);  // -> [0,1]
        float t  = x * nf;
        float fm = floorf(t);                                   // pmin
        pr[d] = (x - fm * wd) * nf;                             // frac pos in cell
        om[d] = 1.0f - pr[d];
        unsigned c = (unsigned)fm;
        h0[d] = c * primes[d];
        h1[d] = h0[d] + primes[d];
    }

    // Fold level into the row index: fidx = l*T + idx = (l<<19)|idx (< 2^23),
    // so every gather is saddr(uniform base) + 32-bit voffset, scale_offset.
    const v2f* __restrict__ tp    = (const v2f*)tab;
    const unsigned          lbase = l << 19;

    float a0 = 0.0f, a1 = 0.0f;
    #pragma unroll
    for (int c = 0; c < 8; ++c) {
        const int bx = (c >> 2) & 1, by = (c >> 1) & 1, bz = c & 1;
        unsigned idx = ((bx ? h1[0] : h0[0]) ^
                        (by ? h1[1] : h0[1]) ^
                        (bz ? h1[2] : h0[2])) & kMask;
        v2f f = tp[lbase | idx];                                 // b64 gather (L2-resident)
        float w = (bx ? pr[0] : om[0]) *
                  (by ? pr[1] : om[1]) *
                  (bz ? pr[2] : om[2]);
        a0 = fmaf(w, f.x, a0);
        a1 = fmaf(w, f.y, a1);
    }

    // Coalesced (256B/wave), non-temporal: keep the 64MB table resident in L2.
    v2f r = {a0, a1};
    __builtin_nontemporal_store(r, (v2f*)out + ((size_t)b * kL + l));
}

extern "C" void kernel_launch(void* const* d_in, const int* in_sizes, int n_in,
                              void* d_out, int out_size, void* d_ws, size_t ws_size,
                              hipStream_t stream) {
    const float* pos = (const float*)d_in[0];   // [B,3]  float32
    const float* tab = (const float*)d_in[1];   // [L,T,F] float32
    float*       out = (float*)d_out;           // [B,L*F] float32

    const int threads = 256;
    const int blocks  = kB * kL / threads;      // 32768, exact
    hashgrid_encode_kernel<<<blocks, threads, 0, stream>>>(pos, tab, out);
}